// GraphSAGE_62955630624874
// MI455X (gfx1250) — compile-verified
//
#include <hip/hip_runtime.h>

typedef float v2f __attribute__((ext_vector_type(2)));
typedef float v8f __attribute__((ext_vector_type(8)));

// ---------------------------------------------------------------------------
// Zero a float buffer (grid-stride).
// ---------------------------------------------------------------------------
__global__ void zero_f32(float* __restrict__ p, long long n) {
  long long i = (long long)blockIdx.x * blockDim.x + threadIdx.x;
  long long stride = (long long)gridDim.x * blockDim.x;
  for (; i < n; i += stride) p[i] = 0.0f;
}

// ---------------------------------------------------------------------------
// Scatter-sum: for each edge e, agg[dst[e]][:] += feat[src[e]][:]
// One thread handles one (edge, 4-channel chunk). chunksLog2 = log2(C/4).
// Optionally accumulates degree (once per edge, chunk 0).
// ---------------------------------------------------------------------------
__global__ void scatter_accum(const float* __restrict__ feat,
                              const int* __restrict__ src,
                              const int* __restrict__ dst,
                              float* __restrict__ agg,
                              float* __restrict__ deg,
                              long long total,   // n_edges * (C/4)
                              int C,
                              int chunksLog2,
                              int add_deg) {
  long long gid = (long long)blockIdx.x * blockDim.x + threadIdx.x;
  if (gid >= total) return;
  int e  = (int)(gid >> chunksLog2);
  int ch = (int)(gid & ((1 << chunksLog2) - 1));
  int s = src[e];
  int d = dst[e];
  const float4 v = *(const float4*)(feat + (long long)s * C + ch * 4);
  float* ap = agg + (long long)d * C + ch * 4;
  atomicAdd(ap + 0, v.x);
  atomicAdd(ap + 1, v.y);
  atomicAdd(ap + 2, v.z);
  atomicAdd(ap + 3, v.w);
  if (add_deg && ch == 0) atomicAdd(deg + d, 1.0f);
}

// ---------------------------------------------------------------------------
// Fused SAGE layer GEMM:
//   out[m, 0:64] = act( (agg[m,:] * rdeg[m]) @ wl  +  xin[m,:] @ wr  + bias )
// One wave computes a 16x64 output tile with v_wmma_f32_16x16x4_f32.
// K must be a multiple of 4; n must be a multiple of 16.
//
// Fragment layouts (CDNA5 ISA 7.12.2, 32-bit, wave32):
//   A (16x4, MxK): lanes 0-15 -> row=lane, {K=0,K=1}; lanes 16-31 -> {K=2,K=3}
//   B (4x16, KxN): lanes 0-15 -> col=lane, {K=0,K=1}; lanes 16-31 -> {K=2,K=3}
//   C/D (16x16):   VGPR v: lanes 0-15 -> M=v, N=lane; lanes 16-31 -> M=v+8
// ---------------------------------------------------------------------------
__global__ void sage_gemm(const float* __restrict__ xin,   // [n, K]
                          const float* __restrict__ agg,   // [n, K]
                          const float* __restrict__ deg,   // [n]
                          const float* __restrict__ wl,    // [K, 64]
                          const float* __restrict__ wr,    // [K, 64]
                          const float* __restrict__ bias,  // [64]
                          float* __restrict__ out,         // [n, 64]
                          int n, int K) {
  const int wavesPerBlock = blockDim.x >> 5;
  const int tile   = blockIdx.x * wavesPerBlock + (threadIdx.x >> 5);
  const int ntiles = n >> 4;
  if (tile >= ntiles) return;            // wave-uniform: EXEC stays all-ones

  const int lane  = threadIdx.x & 31;
  const int row   = lane & 15;           // A row / B col / D col within tile
  const int khalf = (lane >> 4) << 1;    // 0 for lanes 0-15, 2 for lanes 16-31
  const int m0    = tile << 4;

  const float* xrow = xin + (long long)(m0 + row) * K;
  const float* arow = agg + (long long)(m0 + row) * K;
  const float  rd   = 1.0f / fmaxf(deg[m0 + row], 1.0f);

  v8f c[4] = {};   // 4 N-chunks of 16 -> 16x64 tile

  for (int k0 = 0; k0 < K; k0 += 4) {
    // A fragments: mean-aggregated neighbors (scaled) and self features
    v2f am, ax;
    am.x = arow[k0 + khalf]     * rd;
    am.y = arow[k0 + khalf + 1] * rd;
    ax.x = xrow[k0 + khalf];
    ax.y = xrow[k0 + khalf + 1];

    // B fragments: two adjacent K-rows of each weight matrix
    const float* wl0 = wl + (long long)(k0 + khalf) * 64;
    const float* wl1 = wl0 + 64;
    const float* wr0 = wr + (long long)(k0 + khalf) * 64;
    const float* wr1 = wr0 + 64;

    v2f bl[4], br[4];
#pragma unroll
    for (int j = 0; j < 4; ++j) {
      const int col = j * 16 + row;
      bl[j].x = wl0[col];  bl[j].y = wl1[col];
      br[j].x = wr0[col];  br[j].y = wr1[col];
    }

#pragma unroll
    for (int j = 0; j < 4; ++j) {
      c[j] = __builtin_amdgcn_wmma_f32_16x16x4_f32(
          false, am, false, bl[j], (short)0, c[j], false, false);
      c[j] = __builtin_amdgcn_wmma_f32_16x16x4_f32(
          false, ax, false, br[j], (short)0, c[j], false, false);
    }
  }

  // Writeback: bias + ReLU. D layout: VGPR v -> row m0+v (lanes 0-15) or
  // m0+v+8 (lanes 16-31); column = j*16 + (lane&15).
  const int rbase = m0 + ((lane >> 4) << 3);
  const int ncol  = lane & 15;
#pragma unroll
  for (int j = 0; j < 4; ++j) {
    const float bj = bias[j * 16 + ncol];
#pragma unroll
    for (int v = 0; v < 8; ++v) {
      float val = c[j][v] + bj;
      val = fmaxf(val, 0.0f);
      out[(long long)(rbase + v) * 64 + j * 16 + ncol] = val;
    }
  }
}

// ---------------------------------------------------------------------------
// Host-side launcher
// ---------------------------------------------------------------------------
extern "C" void kernel_launch(void* const* d_in, const int* in_sizes, int n_in,
                              void* d_out, int out_size, void* d_ws, size_t ws_size,
                              hipStream_t stream) {
  const int IN_CH = 128;
  const int HID   = 64;

  const float* x   = (const float*)d_in[0];
  const int*   ei  = (const int*)d_in[1];
  const float* w1l = (const float*)d_in[2];
  const float* w1r = (const float*)d_in[3];
  const float* b1  = (const float*)d_in[4];
  const float* w2l = (const float*)d_in[5];
  const float* w2r = (const float*)d_in[6];
  const float* b2  = (const float*)d_in[7];

  const int n_nodes = in_sizes[0] / IN_CH;
  const int n_edges = in_sizes[1] / 2;
  const int* src = ei;              // edge_index row 0
  const int* dst = ei + n_edges;    // edge_index row 1

  // Workspace layout
  float* agg1 = (float*)d_ws;                         // [n, 128]
  float* deg  = agg1 + (size_t)n_nodes * IN_CH;       // [n]
  float* h    = deg  + (size_t)n_nodes;               // [n, 64]
  float* agg2 = h    + (size_t)n_nodes * HID;         // [n, 64]

  float* outp = (float*)d_out;

  const int ZB = 256;
  // Zero agg1 + deg (contiguous) and agg2
  {
    long long n1 = (long long)n_nodes * (IN_CH + 1);
    zero_f32<<<1024, ZB, 0, stream>>>(agg1, n1);
    long long n2 = (long long)n_nodes * HID;
    zero_f32<<<1024, ZB, 0, stream>>>(agg2, n2);
  }

  // ----- Layer 1 -----
  {
    long long total = (long long)n_edges * (IN_CH / 4);
    int blocks = (int)((total + 255) / 256);
    scatter_accum<<<blocks, 256, 0, stream>>>(x, src, dst, agg1, deg,
                                              total, IN_CH, 5, 1);
    int ntiles = n_nodes / 16;
    int gblocks = (ntiles + 7) / 8;   // 8 waves/block
    sage_gemm<<<gblocks, 256, 0, stream>>>(x, agg1, deg, w1l, w1r, b1,
                                           h, n_nodes, IN_CH);
  }

  // ----- Layer 2 -----
  {
    long long total = (long long)n_edges * (HID / 4);
    int blocks = (int)((total + 255) / 256);
    scatter_accum<<<blocks, 256, 0, stream>>>(h, src, dst, agg2, deg,
                                              total, HID, 4, 0);
    int ntiles = n_nodes / 16;
    int gblocks = (ntiles + 7) / 8;
    sage_gemm<<<gblocks, 256, 0, stream>>>(h, agg2, deg, w2l, w2r, b2,
                                           outp, n_nodes, HID);
  }
}